// HashLevel2D_69372311765525
// MI455X (gfx1250) — compile-verified
//
#include <hip/hip_runtime.h>
#include <stdint.h>

// Binarized triplane hash encoding for MI455X (gfx1250, wave32).
//
// Strategy:
//  1. pack_tables_kernel: the reference binarizes table entries to sign(t)
//     (±1), so only 4 sign bits per table row matter. Pack them into 1 byte
//     per row -> 3 x 512 KiB = 1.5 MiB in d_ws. This shrinks the random
//     gather footprint 16x vs float4 rows and is fully L2-resident
//     (192 MB L2), warmed by the pack pass itself.
//  2. triplane_hash_kernel<true>: 1 thread / point. Coalesced b96 position
//     load, 12 random byte gathers (idx = (x ^ y*PRIME) & 0x7FFFF), then
//     accumulate ±w per feature, nontemporal float4 store of the output.
//
// No WMMA: both operands of the per-point 12x4 reduction vary per point,
// so there is no shared matrix -> no D=AxB mapping; gather rate dominates.

static constexpr uint32_t kTableSize = 524288u;       // 2^19
static constexpr uint32_t kTableMask = kTableSize - 1u;
static constexpr float    kRes       = 512.0f;
static constexpr uint32_t kPrimeY    = 2654435761u;

typedef float v4f __attribute__((ext_vector_type(4)));

__device__ __forceinline__ uint32_t sign_bits4(float4 v) {
  // bit=1 -> feature +1 (t >= 0, incl. -0.0 matching jnp.sign semantics),
  // bit=0 -> feature -1 (t < 0).
  uint32_t b = 0u;
  b |= (v.x >= 0.0f) ? 1u : 0u;
  b |= (v.y >= 0.0f) ? 2u : 0u;
  b |= (v.z >= 0.0f) ? 4u : 0u;
  b |= (v.w >= 0.0f) ? 8u : 0u;
  return b;
}

__global__ __launch_bounds__(256) void pack_tables_kernel(
    const float4* __restrict__ t0,
    const float4* __restrict__ t1,
    const float4* __restrict__ t2,
    uint8_t* __restrict__ packed) {
  uint32_t id = blockIdx.x * 256u + threadIdx.x;
  if (id >= 3u * kTableSize) return;
  uint32_t t = id >> 19;           // which table
  uint32_t r = id & kTableMask;    // row
  const float4* tab = (t == 0u) ? t0 : (t == 1u) ? t1 : t2;
  packed[id] = (uint8_t)sign_bits4(tab[r]);
}

struct PlaneLookup {
  uint32_t idx[4];
  float    w[4];
};

__device__ __forceinline__ PlaneLookup make_plane(float pa, float pb) {
  PlaneLookup pl;
  float sa = pa * kRes;
  float sb = pb * kRes;
  float fa = floorf(sa);
  float fb = floorf(sb);
  float wa = sa - fa;              // fractional weight along first coord
  float wb = sb - fb;
  uint32_t ia = (uint32_t)fa;      // in [0, 511]
  uint32_t ib = (uint32_t)fb;
  uint32_t hb0 = ib * kPrimeY;     // uint32 wraparound == jnp.uint32 mul
  uint32_t hb1 = (ib + 1u) * kPrimeY;
  // offsets order matches reference: (0,0),(0,1),(1,0),(1,1)
  pl.idx[0] = (ia        ^ hb0) & kTableMask;
  pl.idx[1] = (ia        ^ hb1) & kTableMask;
  pl.idx[2] = ((ia + 1u) ^ hb0) & kTableMask;
  pl.idx[3] = ((ia + 1u) ^ hb1) & kTableMask;
  float oa = 1.0f - wa;
  float ob = 1.0f - wb;
  pl.w[0] = oa * ob;
  pl.w[1] = oa * wb;
  pl.w[2] = wa * ob;
  pl.w[3] = wa * wb;
  return pl;
}

template <bool PACKED>
__global__ __launch_bounds__(256) void triplane_hash_kernel(
    const float*  __restrict__ pos,
    const float4* __restrict__ txy,
    const float4* __restrict__ txz,
    const float4* __restrict__ tyz,
    const uint8_t* __restrict__ packed,
    v4f* __restrict__ out,
    uint32_t n) {
  uint32_t i = blockIdx.x * 256u + threadIdx.x;
  if (i >= n) return;

  // Speculative lookahead on the streamed positions (global_prefetch_b8).
  if (i + 8192u < n) __builtin_prefetch(pos + 3u * (i + 8192u), 0, 0);

  // Coalesced 12B position load (compiler merges to global_load_b96).
  float p0 = pos[3u * i + 0u];
  float p1 = pos[3u * i + 1u];
  float p2 = pos[3u * i + 2u];

  PlaneLookup pl[3];
  pl[0] = make_plane(p0, p1);   // xy
  pl[1] = make_plane(p0, p2);   // xz
  pl[2] = make_plane(p1, p2);   // yz

  // Issue all 12 gathers before consuming (maximize MLP).
  uint32_t bits[3][4];
  if (PACKED) {
#pragma unroll
    for (int p = 0; p < 3; ++p) {
      const uint8_t* tp = packed + (uint32_t)p * kTableSize;
#pragma unroll
      for (int c = 0; c < 4; ++c) bits[p][c] = (uint32_t)tp[pl[p].idx[c]];
    }
  } else {
    const float4* tabs[3] = {txy, txz, tyz};
#pragma unroll
    for (int p = 0; p < 3; ++p) {
#pragma unroll
      for (int c = 0; c < 4; ++c) bits[p][c] = sign_bits4(tabs[p][pl[p].idx[c]]);
    }
  }

  float a0 = 0.0f, a1 = 0.0f, a2 = 0.0f, a3 = 0.0f;
#pragma unroll
  for (int p = 0; p < 3; ++p) {
#pragma unroll
    for (int c = 0; c < 4; ++c) {
      float w  = pl[p].w[c];
      float nw = -w;
      uint32_t b = bits[p][c];
      a0 += (b & 1u) ? w : nw;   // v_cndmask + v_add per feature
      a1 += (b & 2u) ? w : nw;
      a2 += (b & 4u) ? w : nw;
      a3 += (b & 8u) ? w : nw;
    }
  }

  v4f o;
  o.x = a0; o.y = a1; o.z = a2; o.w = a3;
  // Streamed output: nontemporal store keeps L2 for the gather tables.
  __builtin_nontemporal_store(o, out + i);
}

extern "C" void kernel_launch(void* const* d_in, const int* in_sizes, int n_in,
                              void* d_out, int out_size, void* d_ws, size_t ws_size,
                              hipStream_t stream) {
  const float*  pos = (const float*)d_in[0];
  const float4* txy = (const float4*)d_in[1];
  const float4* txz = (const float4*)d_in[2];
  const float4* tyz = (const float4*)d_in[3];
  uint32_t n = (uint32_t)(in_sizes[0] / 3);
  v4f* out = (v4f*)d_out;

  dim3 blk(256);
  dim3 grd((n + 255u) / 256u);

  const size_t need = 3ull * (size_t)kTableSize;  // 1.5 MiB of packed sign bits
  if (d_ws != nullptr && ws_size >= need) {
    uint8_t* packed = (uint8_t*)d_ws;
    dim3 pgrd((3u * kTableSize + 255u) / 256u);
    pack_tables_kernel<<<pgrd, blk, 0, stream>>>(txy, txz, tyz, packed);
    triplane_hash_kernel<true><<<grd, blk, 0, stream>>>(
        pos, txy, txz, tyz, packed, out, n);
  } else {
    triplane_hash_kernel<false><<<grd, blk, 0, stream>>>(
        pos, txy, txz, tyz, nullptr, out, n);
  }
}